// MultiMambaBlock_34651796144466
// MI455X (gfx1250) — compile-verified
//
#include <hip/hip_runtime.h>
#include <math.h>

// Mamba block for MI455X (gfx1250): fp32 WMMA (16x16x4) for all four GEMMs.
// Big GEMMs (in-proj, out-proj) use 2x2 register-blocked 32x32 wave tiles to
// double WMMA arithmetic intensity and halve L2 traffic. LDS-staged
// channel-parallel scan, fused conv/SiLU/softplus epilogues.

#define BB 2
#define NN 8
#define LL 1024
#define DD 256
#define DIX 512
#define KC 4

typedef __attribute__((ext_vector_type(2))) float v2f;
typedef __attribute__((ext_vector_type(8))) float v8f;

#if defined(__has_builtin)
#if __has_builtin(__builtin_amdgcn_wmma_f32_16x16x4_f32)
#define HAS_WMMA_F32 1
#endif
#endif
#ifndef HAS_WMMA_F32
#define HAS_WMMA_F32 0
#endif

// Fragment layout (CDNA5 ISA 7.12.2), f32 16x16x4:
//   A 16x4 : lane m=lane&15 holds K = k+2*half, k+2*half+1 in v0,v1
//   B 4x16 : lane n=lane&15 holds rows K = k+2*half, k+2*half+1 in v0,v1
//   C/D    : VGPR r, lanes 0-15 -> M=r, lanes 16-31 -> M=r+8, N=lane&15

// ---- single 16x16 tile (small GEMMs) ----------------------------------------
__device__ __forceinline__ v8f gemm_tile_16x16(const float* __restrict__ A, int lda,
                                               const float* __restrict__ B, int ldb,
                                               int K) {
  v8f acc = {0.f, 0.f, 0.f, 0.f, 0.f, 0.f, 0.f, 0.f};
  const int lane = threadIdx.x & 31;
  const int half = lane >> 4;
  const int idx  = lane & 15;
#if HAS_WMMA_F32
  for (int k = 0; k < K; k += 4) {
    v2f a, b;
    a.x = A[idx * lda + (k + 2 * half + 0)];
    a.y = A[idx * lda + (k + 2 * half + 1)];
    b.x = B[(k + 2 * half + 0) * ldb + idx];
    b.y = B[(k + 2 * half + 1) * ldb + idx];
    acc = __builtin_amdgcn_wmma_f32_16x16x4_f32(false, a, false, b,
                                                (short)0, acc, false, false);
  }
#else
  for (int r = 0; r < 8; ++r) {
    const int m = r + half * 8;
    float s = 0.f;
    for (int k = 0; k < K; ++k) s = fmaf(A[m * lda + k], B[k * ldb + idx], s);
    acc[r] = s;
  }
#endif
  return acc;
}

// ---- 2x2 register-blocked 32x32 tile (big GEMMs) ----------------------------
// c{mi}{ni}: rows += 16*mi, cols += 16*ni relative to (A,B) tile origin.
__device__ __forceinline__ void gemm_tile_32x32(const float* __restrict__ A, int lda,
                                                const float* __restrict__ B, int ldb,
                                                int K, v8f& c00, v8f& c01,
                                                v8f& c10, v8f& c11) {
  const int lane = threadIdx.x & 31;
  const int half = lane >> 4;
  const int idx  = lane & 15;
#if HAS_WMMA_F32
  for (int k = 0; k < K; k += 4) {
    v2f a0, a1, b0, b1;
    const float* Ap = A + idx * lda + (k + 2 * half);
    a0.x = Ap[0];            a0.y = Ap[1];
    a1.x = Ap[16 * lda];     a1.y = Ap[16 * lda + 1];
    const float* Bp = B + (k + 2 * half) * ldb + idx;
    b0.x = Bp[0];            b0.y = Bp[ldb];
    b1.x = Bp[16];           b1.y = Bp[ldb + 16];
    c00 = __builtin_amdgcn_wmma_f32_16x16x4_f32(false, a0, false, b0, (short)0, c00, false, false);
    c01 = __builtin_amdgcn_wmma_f32_16x16x4_f32(false, a0, false, b1, (short)0, c01, false, false);
    c10 = __builtin_amdgcn_wmma_f32_16x16x4_f32(false, a1, false, b0, (short)0, c10, false, false);
    c11 = __builtin_amdgcn_wmma_f32_16x16x4_f32(false, a1, false, b1, (short)0, c11, false, false);
  }
#else
  for (int r = 0; r < 8; ++r) {
    const int m = r + half * 8;
    float s00 = 0.f, s01 = 0.f, s10 = 0.f, s11 = 0.f;
    for (int k = 0; k < K; ++k) {
      const float bA = B[k * ldb + idx], bB = B[k * ldb + idx + 16];
      const float aA = A[m * lda + k], aB = A[(m + 16) * lda + k];
      s00 = fmaf(aA, bA, s00); s01 = fmaf(aA, bB, s01);
      s10 = fmaf(aB, bA, s10); s11 = fmaf(aB, bB, s11);
    }
    c00[r] = s00; c01[r] = s01; c10[r] = s10; c11[r] = s11;
  }
#endif
}

// ---------------- Stage 1: in-projection  x[bn,L,D] @ w_in[n][D,2*DI] --------
// 16 bn * 32 * 32 (32x32 tiles) = 16384 waves; 8 waves/block -> 2048 blocks.
__global__ void k_inproj(const float* __restrict__ x, const float* __restrict__ w_in,
                         float* __restrict__ xh, float* __restrict__ res) {
  const int wid = blockIdx.x * 8 + (threadIdx.x >> 5);
  const int bn = wid >> 10;
  const int t  = wid & 1023;
  const int tm = t >> 5, tn = t & 31;       // 32-row / 32-col tile indices
  const int n  = bn & 7;
  const float* A = x + ((size_t)bn * LL + (size_t)tm * 32) * DD;
  const float* B = w_in + (size_t)n * DD * (2 * DIX) + tn * 32;
  v8f c00 = {0.f,0.f,0.f,0.f,0.f,0.f,0.f,0.f}, c01 = c00, c10 = c00, c11 = c00;
  gemm_tile_32x32(A, DD, B, 2 * DIX, DD, c00, c01, c10, c11);
  const int lane = threadIdx.x & 31;
  const int half = lane >> 4, col = lane & 15;
  // whole 32-col tile lands in one buffer (DI=512 is a multiple of 32)
  float* outb; int cbase;
  if (tn * 32 < DIX) { outb = xh;  cbase = tn * 32; }
  else               { outb = res; cbase = tn * 32 - DIX; }
#pragma unroll
  for (int r = 0; r < 8; ++r) {
    const int r0 = tm * 32 + r + half * 8;
    const size_t o0 = ((size_t)bn * LL + r0) * DIX + cbase + col;
    const size_t o1 = ((size_t)bn * LL + r0 + 16) * DIX + cbase + col;
    outb[o0]      = c00[r];
    outb[o0 + 16] = c01[r];
    outb[o1]      = c10[r];
    outb[o1 + 16] = c11[r];
  }
}

// ---------------- Stage 2: depthwise causal conv (K=4) + bias + SiLU --------
__global__ void k_conv_silu(const float* __restrict__ xh, const float* __restrict__ cw,
                            const float* __restrict__ cb, float* __restrict__ xc) {
  const size_t tid = (size_t)blockIdx.x * blockDim.x + threadIdx.x;
  const int di = (int)(tid & (DIX - 1));
  const int l  = (int)((tid >> 9) & (LL - 1));
  const int bn = (int)(tid >> 19);
  const int n  = bn & 7;
  const int c  = n * DIX + di;
  float s = cb[c];
#pragma unroll
  for (int k = 0; k < KC; ++k) {
    const int lk = l - (KC - 1) + k;
    if (lk >= 0)
      s = fmaf(xh[((size_t)bn * LL + lk) * DIX + di], cw[c * KC + k], s);
  }
  s = s / (1.f + expf(-s));
  xc[tid] = s;
}

// ---------------- Stage 3: x_dbl = xc[bn,L,DI] @ w_x[n][DI,48] ---------------
// 16 bn * 64 * 3 (16x16 tiles) = 3072 waves; 1 wave/block.
__global__ void k_xdbl(const float* __restrict__ xc, const float* __restrict__ w_x,
                       float* __restrict__ dtraw, float* __restrict__ Bmb,
                       float* __restrict__ Cmb) {
  const int wid = blockIdx.x;
  const int bn = wid / 192;
  const int t  = wid % 192;
  const int tm = t / 3, tn = t % 3;
  const int n  = bn & 7;
  const float* A = xc + ((size_t)bn * LL + (size_t)tm * 16) * DIX;
  const float* B = w_x + (size_t)n * DIX * 48 + tn * 16;
  v8f acc = gemm_tile_16x16(A, DIX, B, 48, DIX);
  const int lane = threadIdx.x & 31;
  const int half = lane >> 4, col = lane & 15;
  float* outb = (tn == 0) ? dtraw : (tn == 1 ? Bmb : Cmb);
#pragma unroll
  for (int r = 0; r < 8; ++r) {
    const int row = tm * 16 + r + half * 8;
    outb[((size_t)bn * LL + row) * 16 + col] = acc[r];
  }
}

// ---------------- Stage 4: dt = dtraw @ w_dt[n] + b_dt, softplus -------------
// 16 bn * 64 * 32 (16x16 tiles) = 32768 waves; 8 waves/block -> 4096 blocks.
__global__ void k_dtproj(const float* __restrict__ dtraw, const float* __restrict__ w_dt,
                         const float* __restrict__ b_dt, float* __restrict__ dtf) {
  const int wid = blockIdx.x * 8 + (threadIdx.x >> 5);
  const int bn = wid >> 11;
  const int t  = wid & 2047;
  const int tm = t >> 5, tn = t & 31;
  const int n  = bn & 7;
  const float* A = dtraw + ((size_t)bn * LL + (size_t)tm * 16) * 16;
  const float* B = w_dt + (size_t)n * 16 * DIX + tn * 16;
  v8f acc = gemm_tile_16x16(A, 16, B, DIX, 16);
  const int lane = threadIdx.x & 31;
  const int half = lane >> 4, col = lane & 15;
  const int di = tn * 16 + col;
  const float bias = b_dt[n * DIX + di];
#pragma unroll
  for (int r = 0; r < 8; ++r) {
    const int row = tm * 16 + r + half * 8;
    float v = acc[r] + bias;
    v = fmaxf(v, 0.f) + log1pf(expf(-fabsf(v)));   // stable softplus
    dtf[((size_t)bn * LL + row) * DIX + di] = v;
  }
}

// ---------------- Stage 5: selective scan + SiLU(res) gating -----------------
// One block per (b,n); 512 threads = one di channel each; 16-state recurrence
// in registers; B/C chunks staged in LDS (shared by all channels of the head).
__global__ void __launch_bounds__(512) k_scan(
    const float* __restrict__ dtf, const float* __restrict__ Bmb,
    const float* __restrict__ Cmb, const float* __restrict__ xc,
    const float* __restrict__ A_log, const float* __restrict__ Dp,
    const float* __restrict__ resb, float* __restrict__ y) {
  const int bn = blockIdx.x;
  const int n  = bn & 7;
  const int di = threadIdx.x;
  __shared__ float sB[64 * 16];
  __shared__ float sC[64 * 16];
  float Av[16], h[16];
#pragma unroll
  for (int s = 0; s < 16; ++s) {
    Av[s] = -expf(A_log[((size_t)n * DIX + di) * 16 + s]);
    h[s] = 0.f;
  }
  const float dpar = Dp[n * DIX + di];
  for (int t0 = 0; t0 < LL; t0 += 64) {
    __syncthreads();
    for (int i = threadIdx.x; i < 64 * 16; i += 512) {
      sB[i] = Bmb[((size_t)bn * LL + t0) * 16 + i];
      sC[i] = Cmb[((size_t)bn * LL + t0) * 16 + i];
    }
    __syncthreads();
    for (int tt = 0; tt < 64; ++tt) {
      const size_t base = ((size_t)bn * LL + (t0 + tt)) * DIX + di;
      const float dtv = dtf[base];
      const float u   = xc[base];
      float yv = dpar * u;
#pragma unroll
      for (int s = 0; s < 16; ++s) {
        const float hb = h[s] * expf(dtv * Av[s]) + dtv * sB[tt * 16 + s] * u;
        h[s] = hb;
        yv = fmaf(hb, sC[tt * 16 + s], yv);
      }
      const float rv = resb[base];
      yv *= rv / (1.f + expf(-rv));   // * silu(res)
      y[base] = yv;
    }
  }
}

// ---------------- Stage 6: out = y[bn,L,DI] @ w_out[n][DI,D] -----------------
// 16 bn * 32 * 8 (32x32 tiles) = 4096 waves; 8 waves/block -> 512 blocks.
__global__ void k_outproj(const float* __restrict__ y, const float* __restrict__ w_out,
                          float* __restrict__ out) {
  const int wid = blockIdx.x * 8 + (threadIdx.x >> 5);
  const int bn = wid >> 8;
  const int t  = wid & 255;
  const int tm = t >> 3, tn = t & 7;        // 32-row / 32-col tile indices
  const int n  = bn & 7;
  const float* A = y + ((size_t)bn * LL + (size_t)tm * 32) * DIX;
  const float* B = w_out + (size_t)n * DIX * DD + tn * 32;
  v8f c00 = {0.f,0.f,0.f,0.f,0.f,0.f,0.f,0.f}, c01 = c00, c10 = c00, c11 = c00;
  gemm_tile_32x32(A, DIX, B, DD, DIX, c00, c01, c10, c11);
  const int lane = threadIdx.x & 31;
  const int half = lane >> 4, col = lane & 15;
#pragma unroll
  for (int r = 0; r < 8; ++r) {
    const int r0 = tm * 32 + r + half * 8;
    const size_t o0 = ((size_t)bn * LL + r0) * DD + tn * 32 + col;
    const size_t o1 = ((size_t)bn * LL + r0 + 16) * DD + tn * 32 + col;
    out[o0]      = c00[r];
    out[o0 + 16] = c01[r];
    out[o1]      = c10[r];
    out[o1 + 16] = c11[r];
  }
}

extern "C" void kernel_launch(void* const* d_in, const int* in_sizes, int n_in,
                              void* d_out, int out_size, void* d_ws, size_t ws_size,
                              hipStream_t stream) {
  const float* x      = (const float*)d_in[0];
  const float* w_in   = (const float*)d_in[1];
  const float* conv_w = (const float*)d_in[2];
  const float* conv_b = (const float*)d_in[3];
  const float* w_x    = (const float*)d_in[4];
  const float* w_dt   = (const float*)d_in[5];
  const float* b_dt   = (const float*)d_in[6];
  const float* A_log  = (const float*)d_in[7];
  const float* D_p    = (const float*)d_in[8];
  const float* w_out  = (const float*)d_in[9];
  float* out = (float*)d_out;

  float* ws = (float*)d_ws;
  const size_t BIG = (size_t)BB * NN * LL * DIX;   // 8,388,608 floats
  const size_t SML = (size_t)BB * NN * LL * 16;    //   262,144 floats
  float* xh    = ws;               // stage1 out / reused as y (stage5 out)
  float* res   = ws + BIG;
  float* xc    = ws + 2 * BIG;
  float* dtf   = ws + 3 * BIG;
  float* dtraw = ws + 4 * BIG;
  float* Bmb   = dtraw + SML;
  float* Cmb   = Bmb + SML;
  float* y     = xh;               // xh fully consumed by k_conv_silu

  k_inproj   <<<2048, 256, 0, stream>>>(x, w_in, xh, res);
  k_conv_silu<<<(int)(BIG / 256), 256, 0, stream>>>(xh, conv_w, conv_b, xc);
  k_xdbl     <<<3072, 32, 0, stream>>>(xc, w_x, dtraw, Bmb, Cmb);
  k_dtproj   <<<4096, 256, 0, stream>>>(dtraw, w_dt, b_dt, dtf);
  k_scan     <<<16, 512, 0, stream>>>(dtf, Bmb, Cmb, xc, A_log, D_p, res, y);
  k_outproj  <<<512, 256, 0, stream>>>(y, w_out, out);
}